// GAM_Module_20160576487498
// MI455X (gfx1250) — compile-verified
//
#include <hip/hip_runtime.h>
#include <hip/hip_bf16.h>
#include <math.h>

typedef __attribute__((ext_vector_type(2))) float v2f;
typedef __attribute__((ext_vector_type(8))) float v8f;
typedef unsigned int v4u __attribute__((ext_vector_type(4)));
typedef int v8i __attribute__((ext_vector_type(8)));
typedef int v4i __attribute__((ext_vector_type(4)));

#define D_DIM 512
#define N_DIM 16384 /* K*S = 32*512 */
#define B_DIM 8

__device__ inline v8f wmma_f32x4(v2f a, v2f b, v8f c) {
  // (neg_a, A, neg_b, B, c_mod, C, reuse_a, reuse_b)
  return __builtin_amdgcn_wmma_f32_16x16x4_f32(false, a, false, b, (short)0, c,
                                               false, false);
}

// Low 32 bits of a generic LDS pointer == byte offset into the workgroup's
// LDS allocation (ISA 10.2: LDS flat addresses carry offset in addr[31:0]).
static __device__ inline unsigned lds_off(const void* p) {
  return (unsigned)(unsigned long long)(uintptr_t)p;
}

// TDM 2D tile load: global (row_stride_elems fp32 per row) -> LDS, row-major,
// with LDS row padding done by the TDM pad engine.
// pad_interval_code: 0=2,1=4,2=8,3=16,4=32,5=64,6=128,7=256 DWORDs between pads
// pad_amount_code:   0=1 DWORD of pad, ...
static __device__ inline void tdm_load_2d(unsigned lds_byte, const float* gptr,
                                          unsigned tile_w, unsigned tile_h,
                                          unsigned row_stride_elems,
                                          unsigned pad_interval_code,
                                          unsigned pad_amount_code) {
  const unsigned long long ga = (unsigned long long)(uintptr_t)gptr;
  v4u g0;
  g0[0] = 1u;                                  // count=1 (valid user descriptor)
  g0[1] = lds_byte;                            // lds_addr
  g0[2] = (unsigned)ga;                        // global_addr[31:0]
  g0[3] = (unsigned)((ga >> 32) & 0x01FFFFFFu) // global_addr[56:32]
          | (2u << 30);                        // type = 2 ("image")
  v8i g1;
  g1[0] = (int)((2u << 16)                     // data_size = 4 bytes
                | (1u << 20)                   // pad_enable
                | (pad_interval_code << 22) | (pad_amount_code << 25));
  const unsigned td = 0x40000000u;             // huge tensor dims: tiles are
                                               // always in-bounds, never clip
  g1[1] = (int)((td & 0xFFFFu) << 16);         // tensor_dim0[15:0]
  g1[2] = (int)(((td >> 16) & 0xFFFFu)         // tensor_dim0[31:16]
                | ((td & 0xFFFFu) << 16));     // tensor_dim1[15:0]
  g1[3] = (int)(((td >> 16) & 0xFFFFu)         // tensor_dim1[31:16]
                | ((tile_w & 0xFFFFu) << 16)); // tile_dim0
  g1[4] = (int)(tile_h & 0xFFFFu);             // tile_dim1 (tile_dim2 = 0)
  g1[5] = (int)row_stride_elems;               // tensor_dim0_stride[31:0]
  g1[6] = 0;                                   // stride hi / dim1_stride lo
  g1[7] = 0;
  v4i z4 = {0, 0, 0, 0};
#if __clang_major__ >= 23
  v8i z8 = {0, 0, 0, 0, 0, 0, 0, 0};
  __builtin_amdgcn_tensor_load_to_lds(g0, g1, z4, z4, z8, 0);
#else
  __builtin_amdgcn_tensor_load_to_lds(g0, g1, z4, z4, 0);
#endif
}

// =====================  Phase 1: E_b = X_b^T X_b  =====================
// X_b: [N_DIM, D_DIM] row-major. Block = 256 threads (8 waves).
// Output tile: 64 (E rows) x 128 (E cols). Wave w: tile-row = w>>1,
// tile-cols = (w&1)*64 .. +63 (4 accumulator tiles).
// K-slabs of 32 rows, double-buffered TDM staging.
#define P1_BM 64
#define P1_BN 128
#define P1_KB 32

__global__ __launch_bounds__(256) void gram_kernel(const float* __restrict__ x,
                                                   float* __restrict__ E) {
  const int bid = blockIdx.x;
  const int b = bid >> 5; // 32 blocks per batch
  const int t = bid & 31;
  const int row0 = (t >> 2) * P1_BM;
  const int col0 = (t & 3) * P1_BN;
  const float* X = x + (size_t)b * N_DIM * D_DIM;

  __shared__ float sA[2][P1_KB][P1_BM + 1]; // sA[q][kk][c] = X[k0+kk, row0+c]
  __shared__ float sB[2][P1_KB][P1_BN + 1]; // sB[q][kk][c] = X[k0+kk, col0+c]

  const int tid = threadIdx.x;
  const int lane = tid & 31;
  const int w = tid >> 5;
  const int m = lane & 15;
  const int kg = (lane >> 4) << 1; // 0 (lanes 0-15) / 2 (lanes 16-31)
  const int trow = (w >> 1) * 16;
  const int tcol0 = (w & 1) * 64;

  v8f acc[4];
  const v8f vzero = {0.f, 0.f, 0.f, 0.f, 0.f, 0.f, 0.f, 0.f};
  acc[0] = vzero; acc[1] = vzero; acc[2] = vzero; acc[3] = vzero;

  if (w == 0) { // prologue: slab 0 into buffer 0
    tdm_load_2d(lds_off(&sA[0][0][0]), X + row0, P1_BM, P1_KB, D_DIM, 5, 0);
    tdm_load_2d(lds_off(&sB[0][0][0]), X + col0, P1_BN, P1_KB, D_DIM, 6, 0);
  }

  int cur = 0;
  for (int k0 = 0; k0 < N_DIM; k0 += P1_KB) {
    if (w == 0) __builtin_amdgcn_s_wait_tensorcnt(0);
    __syncthreads(); // slab `cur` landed; everyone done with `cur^1`
    if (w == 0 && (k0 + P1_KB) < N_DIM) { // prefetch next slab via TDM
      const float* nx = X + (size_t)(k0 + P1_KB) * D_DIM;
      tdm_load_2d(lds_off(&sA[cur ^ 1][0][0]), nx + row0, P1_BM, P1_KB, D_DIM, 5, 0);
      tdm_load_2d(lds_off(&sB[cur ^ 1][0][0]), nx + col0, P1_BN, P1_KB, D_DIM, 6, 0);
    }
#pragma unroll
    for (int kk = 0; kk < P1_KB; kk += 4) {
      v2f a;
      a.x = sA[cur][kk + kg][trow + m];
      a.y = sA[cur][kk + kg + 1][trow + m];
#pragma unroll
      for (int tc = 0; tc < 4; ++tc) {
        v2f bf;
        bf.x = sB[cur][kk + kg][tcol0 + tc * 16 + m];
        bf.y = sB[cur][kk + kg + 1][tcol0 + tc * 16 + m];
        acc[tc] = wmma_f32x4(a, bf, acc[tc]);
      }
    }
    cur ^= 1;
  }

  // C/D layout: VGPR r -> M=r (lanes 0-15) / M=8+r (lanes 16-31); N = lane&15
  float* Eb = E + (size_t)b * D_DIM * D_DIM;
  const int mhalf = (lane >> 4) << 3;
#pragma unroll
  for (int tc = 0; tc < 4; ++tc) {
    const int col = col0 + tcol0 + tc * 16 + m;
#pragma unroll
    for (int r = 0; r < 8; ++r) {
      const int row = row0 + trow + mhalf + r;
      Eb[(size_t)row * D_DIM + col] = acc[tc][r];
    }
  }
}

// ==============  Phase 2: A = softmax(rowmax(E) - E) in-place  ==============
// softmax(rowmax - e) == softmax(-e); stable form: exp(rowmin - e) / sum.
__global__ __launch_bounds__(256) void softmax_kernel(float* __restrict__ E) {
  const int tid = threadIdx.x;
  const int lane = tid & 31;
  const int w = tid >> 5;
  const int row = blockIdx.x * 8 + w; // [0, B*D)
  float* Erow = E + (size_t)row * D_DIM;

  float vals[16];
  float rmin = 3.4e38f;
#pragma unroll
  for (int i = 0; i < 16; ++i) {
    vals[i] = Erow[lane + i * 32];
    rmin = fminf(rmin, vals[i]);
  }
#pragma unroll
  for (int o = 16; o >= 1; o >>= 1) rmin = fminf(rmin, __shfl_xor(rmin, o, 32));
  float sum = 0.f;
#pragma unroll
  for (int i = 0; i < 16; ++i) {
    vals[i] = __expf(rmin - vals[i]);
    sum += vals[i];
  }
#pragma unroll
  for (int o = 16; o >= 1; o >>= 1) sum += __shfl_xor(sum, o, 32);
  const float inv = 1.0f / sum;
#pragma unroll
  for (int i = 0; i < 16; ++i) Erow[lane + i * 32] = vals[i] * inv;
}

// ==============  Phase 3: Y = gamma * (X A^T) + X  ==============
// Output tile per block: 128 (n rows) x 64 (d cols). Wave w owns n-tile-row w
// and 4 d-tiles. LDS tiles are natural row-major [row][k] (TDM-friendly),
// rows padded to 33 floats by the TDM pad engine.
#define P3_BM 128
#define P3_BN 64
#define P3_KB 32

__global__ __launch_bounds__(256) void out_kernel(const float* __restrict__ x,
                                                  const float* __restrict__ att,
                                                  const float* __restrict__ gamma,
                                                  float* __restrict__ y) {
  const int bid = blockIdx.x;
  const int bpb = (N_DIM / P3_BM) * (D_DIM / P3_BN); // 128*8 = 1024 per batch
  const int b = bid / bpb;
  const int t = bid % bpb;
  const int n0 = (t >> 3) * P3_BM;
  const int d0 = (t & 7) * P3_BN;

  const float* Xb = x + (size_t)b * N_DIM * D_DIM;
  const float* Ab = att + (size_t)b * D_DIM * D_DIM;
  float* Yb = y + (size_t)b * N_DIM * D_DIM;

  __shared__ float sX[2][P3_BM][P3_KB + 1]; // sX[q][r][kk] = X[n0+r, k0+kk]
  __shared__ float sT[2][P3_BN][P3_KB + 1]; // sT[q][c][kk] = att[d0+c, k0+kk]

  const int tid = threadIdx.x;
  const int lane = tid & 31;
  const int w = tid >> 5;
  const int m = lane & 15;
  const int kg = (lane >> 4) << 1;

  v8f acc[4];
  const v8f vzero = {0.f, 0.f, 0.f, 0.f, 0.f, 0.f, 0.f, 0.f};
  acc[0] = vzero; acc[1] = vzero; acc[2] = vzero; acc[3] = vzero;

  if (w == 0) { // prologue: slab 0 into buffer 0
    tdm_load_2d(lds_off(&sX[0][0][0]), Xb + n0 * D_DIM, P3_KB, P3_BM, D_DIM, 4, 0);
    tdm_load_2d(lds_off(&sT[0][0][0]), Ab + d0 * D_DIM, P3_KB, P3_BN, D_DIM, 4, 0);
  }

  int cur = 0;
  for (int k0 = 0; k0 < D_DIM; k0 += P3_KB) {
    if (w == 0) __builtin_amdgcn_s_wait_tensorcnt(0);
    __syncthreads();
    if (w == 0 && (k0 + P3_KB) < D_DIM) {
      tdm_load_2d(lds_off(&sX[cur ^ 1][0][0]), Xb + n0 * D_DIM + k0 + P3_KB,
                  P3_KB, P3_BM, D_DIM, 4, 0);
      tdm_load_2d(lds_off(&sT[cur ^ 1][0][0]), Ab + d0 * D_DIM + k0 + P3_KB,
                  P3_KB, P3_BN, D_DIM, 4, 0);
    }
#pragma unroll
    for (int kk = 0; kk < P3_KB; kk += 4) {
      v2f a;
      a.x = sX[cur][w * 16 + m][kk + kg];
      a.y = sX[cur][w * 16 + m][kk + kg + 1];
#pragma unroll
      for (int tc = 0; tc < 4; ++tc) {
        v2f bf;
        bf.x = sT[cur][tc * 16 + m][kk + kg];
        bf.y = sT[cur][tc * 16 + m][kk + kg + 1];
        acc[tc] = wmma_f32x4(a, bf, acc[tc]);
      }
    }
    cur ^= 1;
  }

  const float g = gamma[0];
  const int mhalf = (lane >> 4) << 3;
#pragma unroll
  for (int tc = 0; tc < 4; ++tc) {
    const int col = d0 + tc * 16 + m;
#pragma unroll
    for (int r = 0; r < 8; ++r) {
      const int row = n0 + w * 16 + mhalf + r;
      const size_t idx = (size_t)row * D_DIM + col;
      Yb[idx] = g * acc[tc][r] + Xb[idx];
    }
  }
}

extern "C" void kernel_launch(void* const* d_in, const int* in_sizes, int n_in,
                              void* d_out, int out_size, void* d_ws, size_t ws_size,
                              hipStream_t stream) {
  const float* x = (const float*)d_in[0];
  const float* gamma = (const float*)d_in[1];
  float* out = (float*)d_out;
  float* E = (float*)d_ws; // B*D*D floats = 8 MB; fully overwritten each call

  gram_kernel<<<B_DIM * 32, 256, 0, stream>>>(x, E);
  softmax_kernel<<<(B_DIM * D_DIM) / 8, 256, 0, stream>>>(E);
  out_kernel<<<B_DIM * 1024, 256, 0, stream>>>(x, E, gamma, out);
}